// GCN_2_12850542150399
// MI455X (gfx1250) — compile-verified
//
#include <hip/hip_runtime.h>

typedef __attribute__((ext_vector_type(2))) float v2f;
typedef __attribute__((ext_vector_type(8))) float v8f;

#define NHID 128

// ---------------------------------------------------------------------------
// fp32 WMMA GEMM: C[M,N] = A[M,K] * B[K,N] (+ bias[N] if bias != nullptr)
// One wave computes a 64x16 tile (4 m-tiles) via V_WMMA_F32_16X16X4_F32.
// B fragment is loaded once per K-step and reused by 4 WMMAs (4x amortized);
// the 4 accumulator chains are independent, hiding the WMMA->WMMA RAW hazard.
// A-fragment layout (ISA 7.12.2, 32-bit A 16x4): lane<16 -> K = k0+{0,1},
// lane>=16 -> K = k0+{2,3}; M = lane&15  => contiguous float2 per lane.
// C/D layout: VGPR r -> row (r + 8*(lane>>4)), col = lane&15.
// ---------------------------------------------------------------------------
__launch_bounds__(32)
__global__ void gcn_gemm_wmma(const float* __restrict__ A,
                              const float* __restrict__ B,
                              const float* __restrict__ bias,
                              float* __restrict__ C,
                              int M, int K, int N) {
    const int lane = threadIdx.x & 31;
    const int half = lane >> 4;
    const int l16  = lane & 15;
    const int mt0  = blockIdx.x * 4;          // 4 consecutive 16-row tiles
    const int nt   = blockIdx.y;
    const int col  = nt * 16 + l16;

    const float* arow[4];
#pragma unroll
    for (int t = 0; t < 4; ++t) {
        int row = (mt0 + t) * 16 + l16;
        if (row >= M) row = M - 1;            // clamp: EXEC must stay all-ones
        arow[t] = A + (size_t)row * K;
    }
    const float* bcol = B + col;

    v8f acc[4] = {};
    for (int k0 = 0; k0 < K; k0 += 8) {
        const int ka = k0 + 2 * half;
        const int kb = ka + 4;
        // B fragment: shared by all 4 m-tiles this K-step
        v2f b0; b0.x = bcol[(size_t)ka * N];
                b0.y = bcol[(size_t)(ka + 1) * N];
        v2f b1; b1.x = bcol[(size_t)kb * N];
                b1.y = bcol[(size_t)(kb + 1) * N];
        // A fragments: contiguous float2 per lane
        v2f a0[4], a1[4];
#pragma unroll
        for (int t = 0; t < 4; ++t) {
            __builtin_prefetch(arow[t] + k0 + 64, 0, 0);   // global_prefetch_b8
            a0[t].x = arow[t][ka];  a0[t].y = arow[t][ka + 1];
            a1[t].x = arow[t][kb];  a1[t].y = arow[t][kb + 1];
        }
        // 4 independent chains, then their dependent partners
#pragma unroll
        for (int t = 0; t < 4; ++t)
            acc[t] = __builtin_amdgcn_wmma_f32_16x16x4_f32(false, a0[t], false, b0,
                                                           (short)0, acc[t], false, false);
#pragma unroll
        for (int t = 0; t < 4; ++t)
            acc[t] = __builtin_amdgcn_wmma_f32_16x16x4_f32(false, a1[t], false, b1,
                                                           (short)0, acc[t], false, false);
    }

    const float bv = bias ? bias[col] : 0.0f;
#pragma unroll
    for (int t = 0; t < 4; ++t) {
        const int rbase = (mt0 + t) * 16 + 8 * half;
#pragma unroll
        for (int r = 0; r < 8; ++r) {
            const int orow = rbase + r;
            if (orow < M)
                C[(size_t)orow * N + col] = acc[t][r] + bv;
        }
    }
}

// ---------------------------------------------------------------------------
// degree / normalization
// ---------------------------------------------------------------------------
__global__ void deg_init(int* __restrict__ deg, int n) {
    int i = blockIdx.x * blockDim.x + threadIdx.x;
    if (i < n) deg[i] = 1;                       // self-loop contribution
}

__global__ void deg_count(const int* __restrict__ dst, int* __restrict__ deg, int e) {
    int i = blockIdx.x * blockDim.x + threadIdx.x;
    if (i < e) atomicAdd(&deg[dst[i]], 1);
}

__global__ void deg_rsqrt(const int* __restrict__ deg, float* __restrict__ dinv, int n) {
    int i = blockIdx.x * blockDim.x + threadIdx.x;
    if (i < n) dinv[i] = rsqrtf((float)deg[i]);  // deg >= 1 always (self-loop)
}

// ---------------------------------------------------------------------------
// agg[i,f] = h[i,f] * dinv[i]^2 + bias[f]   (self-loop edge + GCN bias)
// ---------------------------------------------------------------------------
__global__ void agg_init(const float* __restrict__ h, const float* __restrict__ dinv,
                         const float* __restrict__ bias, float* __restrict__ agg,
                         int n) {
    int t = blockIdx.x * blockDim.x + threadIdx.x;
    if (t >= n * NHID) return;
    int node = t >> 7;            // / NHID
    int f    = t & (NHID - 1);
    float d = dinv[node];
    agg[t] = h[t] * d * d + bias[f];
}

// ---------------------------------------------------------------------------
// One wave per edge: 32 lanes x float4 = 128 features, coalesced.
// agg[dst,:] += h[src,:] * dinv[src]*dinv[dst]   via global_atomic_add_f32.
// h and agg are L2-resident (51 MB each << 192 MB L2).
// ---------------------------------------------------------------------------
__launch_bounds__(256)
__global__ void edge_scatter(const int* __restrict__ src, const int* __restrict__ dst,
                             const float* __restrict__ dinv,
                             const float* __restrict__ h, float* __restrict__ agg,
                             int e) {
    int wave = (blockIdx.x * blockDim.x + threadIdx.x) >> 5;
    int lane = threadIdx.x & 31;
    if (wave >= e) return;
    int s = src[wave];
    int d = dst[wave];
    float nrm = dinv[s] * dinv[d];
    const float4 v = ((const float4*)(h + (size_t)s * NHID))[lane];
    float* ad = agg + (size_t)d * NHID + lane * 4;
    atomicAdd(ad + 0, v.x * nrm);
    atomicAdd(ad + 1, v.y * nrm);
    atomicAdd(ad + 2, v.z * nrm);
    atomicAdd(ad + 3, v.w * nrm);
}

// ---------------------------------------------------------------------------
// launch
// ---------------------------------------------------------------------------
extern "C" void kernel_launch(void* const* d_in, const int* in_sizes, int n_in,
                              void* d_out, int out_size, void* d_ws, size_t ws_size,
                              hipStream_t stream) {
    const float* x     = (const float*)d_in[0];   // [N, 256]
    const int*   ei    = (const int*)d_in[1];     // [2, E]
    // d_in[2] = adj (unused)
    const float* gc1_w = (const float*)d_in[3];   // [256, 128]
    const float* gc1_b = (const float*)d_in[4];   // [128]
    const float* fc_w  = (const float*)d_in[5];   // [128, 128]
    const float* fc_b  = (const float*)d_in[6];   // [128]
    float*       y     = (float*)d_out;           // [N, 128]

    const int NFEAT = 256;
    const int E = in_sizes[1] / 2;
    const int N = in_sizes[0] / NFEAT;            // 100000

    const int* src = ei;
    const int* dst = ei + E;

    // workspace carve-out: h [N,128] | agg [N,128] | deg [N] | dinv [N]
    float* h    = (float*)d_ws;
    float* agg  = h + (size_t)N * NHID;
    int*   deg  = (int*)(agg + (size_t)N * NHID);
    float* dinv = (float*)(deg + N);

    // 1) degree (incl. self-loops) and deg^-1/2
    deg_init <<<(N + 255) / 256, 256, 0, stream>>>(deg, N);
    deg_count<<<(E + 255) / 256, 256, 0, stream>>>(dst, deg, E);
    deg_rsqrt<<<(N + 255) / 256, 256, 0, stream>>>(deg, dinv, N);

    // 2) h = x @ gc1_w   (WMMA fp32, 64x16 tile per wave)
    {
        dim3 g((N + 63) / 64, NHID / 16);
        gcn_gemm_wmma<<<g, 32, 0, stream>>>(x, gc1_w, nullptr, h, N, NFEAT, NHID);
    }

    // 3) agg = h * dinv^2 + gc1_b  (self-loop term + bias)
    agg_init<<<((size_t)N * NHID + 255) / 256, 256, 0, stream>>>(h, dinv, gc1_b, agg, N);

    // 4) scatter-add edge messages
    edge_scatter<<<(E + 7) / 8, 256, 0, stream>>>(src, dst, dinv, h, agg, E);

    // 5) y = agg @ fc_w + fc_b   (WMMA fp32, fused bias)
    {
        dim3 g((N + 63) / 64, NHID / 16);
        gcn_gemm_wmma<<<g, 32, 0, stream>>>(agg, fc_w, fc_b, y, N, NHID, NHID);
    }
}